// MPS_79577154060541
// MI455X (gfx1250) — compile-verified
//
#include <hip/hip_runtime.h>

// MPS chain contraction for MI455X (gfx1250, wave32, WMMA).
// D=16 bond dimension maps exactly onto V_WMMA_F32_16X16X4_F32 tiles.
// Each wave owns one running 16x16 f32 product; no cross-wave coupling.

typedef __attribute__((ext_vector_type(2))) float v2f;
typedef __attribute__((ext_vector_type(8))) float v8f;

#define N_SITES 1024
#define BATCH 256
#define CLS 10
#define CHUNK 32                              // sites per wave (kernel 1)
#define CHUNKS_PER_BATCH (N_SITES / CHUNK)    // 32 (16 per half)
#define WAVES_PER_BLOCK 8                     // 256 threads = 8 wave32

// Wave-local LDS sync: lanes are lockstep in wave32 and LDS ops from one wave
// complete in order (DScnt in-order), so a dscnt wait + compiler memory fence
// replaces the block-wide s_barrier entirely.
#define LDS_SYNC()    asm volatile("s_wait_dscnt 0x0" ::: "memory")
#define SCHED_FENCE() asm volatile("" ::: "memory")

// ---- wave-level 16x16 @ 16x16 f32 product via 4 accumulating WMMAs ----
// A-operand (16x4 f32): lanes 0-15 -> M=lane, VGPR0/1 = K=0/1; lanes 16-31 -> K=2/3.
// B-operand (4x16 f32): VGPR v, lanes 0-15 -> K=v, lanes 16-31 -> K=v+2 (row-striped).
// C/D (16x16 f32): VGPR v, lanes 0-15 -> M=v, lanes 16-31 -> M=v+8; N=lane&15.
__device__ __forceinline__ v8f wmma_mul16(const float (*P)[16], const float (*S)[16], int lane) {
    const int m  = lane & 15;
    const int kh = (lane >> 4) & 1;
    v8f c = {0.f, 0.f, 0.f, 0.f, 0.f, 0.f, 0.f, 0.f};
#pragma unroll
    for (int k = 0; k < 4; ++k) {
        const int base = 4 * k + 2 * kh;
        v2f a, b;
        a.x = P[m][base];       // A = P[:, 4k:4k+4]
        a.y = P[m][base + 1];
        b.x = S[base][m];       // B = S[4k:4k+4, :]
        b.y = S[base + 1][m];
        // 8 args: (neg_a, A, neg_b, B, c_mod, C, reuse_a, reuse_b)
        c = __builtin_amdgcn_wmma_f32_16x16x4_f32(false, a, false, b,
                                                  (short)0, c, false, false);
    }
    return c;
}

__device__ __forceinline__ void store_cd(float (*P)[16], int lane, v8f c) {
    const int n  = lane & 15;
    const int kh = (lane >> 4) & 1;
#pragma unroll
    for (int v = 0; v < 8; ++v) P[v + 8 * kh][n] = c[v];
}

__device__ __forceinline__ void init_identity(float (*P)[16], int lane) {
#pragma unroll
    for (int j = 0; j < 8; ++j) {
        const int e = lane * 8 + j;
        P[e >> 4][e & 15] = ((e >> 4) == (e & 15)) ? 1.0f : 0.0f;
    }
}

// ---- kernel 1: per-(batch, chunk) ordered product of 32 site matrices ----
// Fuses the x·tensor contraction (never materializes the 268 MB A tensor)
// and register-double-buffers the next site's tensor tile so the global
// fetch (LOADcnt) overlaps the 4-deep WMMA chain (independent counters).
__global__ void __launch_bounds__(256)
mps_chunk_kernel(const float* __restrict__ x,       // (N, B, 2)
                 const float* __restrict__ tensor,  // (N, 16, 16, 2)
                 float* __restrict__ chunk_out) {   // (B, 32, 16, 16)
    __shared__ float Pbuf[WAVES_PER_BLOCK][16][16];
    __shared__ float Sbuf[WAVES_PER_BLOCK][16][16];
    const int lane = threadIdx.x & 31;
    const int wv   = threadIdx.x >> 5;
    const int w    = blockIdx.x * WAVES_PER_BLOCK + wv;  // 0..8191
    const int b     = w >> 5;   // batch
    const int chunk = w & 31;   // chunk within batch
    float (*P)[16] = Pbuf[wv];
    float (*S)[16] = Sbuf[wv];

    init_identity(P, lane);

    const int site0 = chunk * CHUNK;

    // Preload site 0 (8 float2 = two b128 clauses per lane + x pair).
    float2 tv[8];
    float xx0, xx1;
    {
        const float2* t = (const float2*)(tensor + (size_t)site0 * 512);
#pragma unroll
        for (int j = 0; j < 8; ++j) tv[j] = t[lane * 8 + j];
        const float* xp = x + ((size_t)site0 * BATCH + b) * 2;
        xx0 = xp[0];
        xx1 = xp[1];
    }
    LDS_SYNC();   // identity P visible to operand loads

    for (int s = 0; s < CHUNK; ++s) {
        // Publish site matrix S = x0*T0 + x1*T1 into this wave's LDS tile.
#pragma unroll
        for (int j = 0; j < 8; ++j) {
            const int e = lane * 8 + j;      // e = row*16 + col
            S[e >> 4][e & 15] = xx0 * tv[j].x + xx1 * tv[j].y;
        }
        LDS_SYNC();

        // Prefetch next site while the WMMA chain runs (LOADcnt vs DScnt).
        if (s + 1 < CHUNK) {
            const int n = site0 + s + 1;
            const float2* t = (const float2*)(tensor + (size_t)n * 512);
#pragma unroll
            for (int j = 0; j < 8; ++j) tv[j] = t[lane * 8 + j];
            const float* xp = x + ((size_t)n * BATCH + b) * 2;
            xx0 = xp[0];
            xx1 = xp[1];
        }

        v8f c = wmma_mul16(P, S, lane);      // P_new = P @ S
        SCHED_FENCE();                       // keep P reads before P writes
        store_cd(P, lane, c);
        LDS_SYNC();                          // P visible for next iteration
    }

    float* o = chunk_out + ((size_t)b * CHUNKS_PER_BATCH + chunk) * 256;
#pragma unroll
    for (int j = 0; j < 8; ++j) {
        const int e = lane * 8 + j;
        o[e] = P[e >> 4][e & 15];
    }
}

// ---- kernel 2: reduce 16 chunk products per (batch, half); extract row0/col0 ----
__global__ void __launch_bounds__(256)
mps_reduce_kernel(const float* __restrict__ chunk_in,  // (B, 32, 16, 16)
                  float* __restrict__ Al,              // (B, 16)
                  float* __restrict__ Ar) {            // (B, 16)
    __shared__ float Pbuf[WAVES_PER_BLOCK][16][16];
    __shared__ float Sbuf[WAVES_PER_BLOCK][16][16];
    const int lane = threadIdx.x & 31;
    const int wv   = threadIdx.x >> 5;
    const int w    = blockIdx.x * WAVES_PER_BLOCK + wv;  // 0..511
    const int b    = w >> 1;
    const int half = w & 1;
    float (*P)[16] = Pbuf[wv];
    float (*S)[16] = Sbuf[wv];

    init_identity(P, lane);

    const float* base = chunk_in + ((size_t)b * CHUNKS_PER_BATCH + half * 16) * 256;

    // Preload chunk matrix 0 (two float4 per lane).
    float4 m0 = ((const float4*)(base))[lane * 2 + 0];
    float4 m1 = ((const float4*)(base))[lane * 2 + 1];
    LDS_SYNC();

    for (int jch = 0; jch < 16; ++jch) {
        const int e0 = lane * 8;
        S[(e0 + 0) >> 4][(e0 + 0) & 15] = m0.x;
        S[(e0 + 1) >> 4][(e0 + 1) & 15] = m0.y;
        S[(e0 + 2) >> 4][(e0 + 2) & 15] = m0.z;
        S[(e0 + 3) >> 4][(e0 + 3) & 15] = m0.w;
        S[(e0 + 4) >> 4][(e0 + 4) & 15] = m1.x;
        S[(e0 + 5) >> 4][(e0 + 5) & 15] = m1.y;
        S[(e0 + 6) >> 4][(e0 + 6) & 15] = m1.z;
        S[(e0 + 7) >> 4][(e0 + 7) & 15] = m1.w;
        LDS_SYNC();

        if (jch + 1 < 16) {
            const float4* nx = (const float4*)(base + (size_t)(jch + 1) * 256);
            m0 = nx[lane * 2 + 0];
            m1 = nx[lane * 2 + 1];
        }

        v8f c = wmma_mul16(P, S, lane);
        SCHED_FENCE();
        store_cd(P, lane, c);
        LDS_SYNC();
    }

    if (lane < 16) {
        if (half == 0) Al[b * 16 + lane] = P[0][lane];   // left boundary: row 0
        else           Ar[b * 16 + lane] = P[lane][0];   // right boundary: col 0
    }
}

// ---- kernel 3: out[b,o] = Al[b] @ Aout[o] @ Ar[b]  (2560 tiny dot products) ----
__global__ void __launch_bounds__(256)
mps_out_kernel(const float* __restrict__ Al, const float* __restrict__ Ar,
               const float* __restrict__ Aout, float* __restrict__ out) {
    const int t = blockIdx.x * blockDim.x + threadIdx.x;  // 0..2559 exactly
    const int b = t / CLS;
    const int o = t % CLS;
    const float* al = Al + b * 16;
    const float* ar = Ar + b * 16;
    const float* A  = Aout + o * 256;
    float acc = 0.f;
#pragma unroll
    for (int l = 0; l < 16; ++l) {
        float s = 0.f;
#pragma unroll
        for (int r = 0; r < 16; ++r) s = fmaf(A[l * 16 + r], ar[r], s);
        acc = fmaf(al[l], s, acc);
    }
    out[t] = acc;
}

extern "C" void kernel_launch(void* const* d_in, const int* in_sizes, int n_in,
                              void* d_out, int out_size, void* d_ws, size_t ws_size,
                              hipStream_t stream) {
    const float* x      = (const float*)d_in[0];  // (1024, 256, 2)
    const float* tensor = (const float*)d_in[1];  // (1024, 16, 16, 2)
    const float* Aout   = (const float*)d_in[2];  // (10, 16, 16)
    float* out = (float*)d_out;                   // (256, 10)

    float* ws     = (float*)d_ws;
    float* chunks = ws;                           // 256*32*256 = 2,097,152 floats (8 MB)
    float* Al     = ws + (size_t)BATCH * CHUNKS_PER_BATCH * 256;  // 4096 floats
    float* Ar     = Al + BATCH * 16;                              // 4096 floats

    // 8192 waves = 256 batches x 32 chunks; 8 waves/block -> 1024 blocks
    mps_chunk_kernel<<<dim3(1024), dim3(256), 0, stream>>>(x, tensor, chunks);
    // 512 waves = 256 batches x 2 halves -> 64 blocks
    mps_reduce_kernel<<<dim3(64), dim3(256), 0, stream>>>(chunks, Al, Ar);
    // 2560 threads -> 10 blocks of 256
    mps_out_kernel<<<dim3(10), dim3(256), 0, stream>>>(Al, Ar, Aout, out);
}